// GAT_16080357556339
// MI455X (gfx1250) — compile-verified
//
#include <hip/hip_runtime.h>
#include <math.h>

typedef float v2f __attribute__((ext_vector_type(2)));
typedef float v8f __attribute__((ext_vector_type(8)));

#define NEG_SLOPE 0.2f

// ---------- monotone float <-> uint encoding for atomicMax-based segment max ----------
__device__ __forceinline__ unsigned enc_f(float f) {
  unsigned u = __float_as_uint(f);
  return (u & 0x80000000u) ? ~u : (u | 0x80000000u);
}
__device__ __forceinline__ float dec_f(unsigned u) {
  return __uint_as_float((u & 0x80000000u) ? (u & 0x7FFFFFFFu) : ~u);
}

// ---------- zero fill ----------
__global__ void zero_kernel(float* __restrict__ p, long long n) {
  long long i = (long long)blockIdx.x * blockDim.x + threadIdx.x;
  long long stride = (long long)gridDim.x * blockDim.x;
  for (; i < n; i += stride) p[i] = 0.0f;
}

// ---------- fp32 WMMA GEMM, 2x2 register-blocked: C[M,Nc] = A[M,K] * B[K,Nc] ----------
// One 32x32 output block per wave32 (four 16x16 WMMA tiles), K-loop in steps of 4 via
// V_WMMA_F32_16X16X4_F32. Per k-step: 2 A loads (b64) + 4 B loads (b32) feed 4 WMMAs.
// A frag: lanes 0-15 hold M=lane,K=k..k+1; lanes 16-31 hold M=lane-16,K=k+2..k+3.
// B frag mirrored; C/D: VGPR i -> (M=i, N=lane) / (M=i+8, N=lane-16).
// Requires M % 32 == 0 and Nc % 32 == 0 (true here: M=100000, Nc=128/32).
__global__ void gemm_wmma_f32_2x2(const float* __restrict__ A, const float* __restrict__ B,
                                  float* __restrict__ C, int M, int K, int Nc) {
  int wave = (blockIdx.x * blockDim.x + threadIdx.x) >> 5;
  int lane = threadIdx.x & 31;
  int nblk = Nc >> 5;                       // 32-wide column blocks
  int bm = wave / nblk;
  int bn = wave - bm * nblk;
  if (bm * 32 >= M) return;                 // wave-uniform: EXEC stays all-1 for WMMA

  int m    = (lane & 15);
  int koff = (lane >> 4) << 1;              // 0 or 2

  const float* arow0 = A + (size_t)(bm * 32 + m)      * K + koff;
  const float* arow1 = A + (size_t)(bm * 32 + 16 + m) * K + koff;
  const float* bcol0 = B + (size_t)(bn * 32 + m);
  const float* bcol1 = B + (size_t)(bn * 32 + 16 + m);

  v8f c00 = {0.f,0.f,0.f,0.f,0.f,0.f,0.f,0.f};
  v8f c01 = c00, c10 = c00, c11 = c00;

#pragma unroll 2
  for (int k = 0; k < K; k += 4) {
    v2f a0 = *(const v2f*)(arow0 + k);
    v2f a1 = *(const v2f*)(arow1 + k);
    v2f b0, b1;
    size_t r0 = (size_t)(k + koff) * Nc;
    size_t r1 = (size_t)(k + koff + 1) * Nc;
    b0.x = bcol0[r0]; b0.y = bcol0[r1];
    b1.x = bcol1[r0]; b1.y = bcol1[r1];
    c00 = __builtin_amdgcn_wmma_f32_16x16x4_f32(false, a0, false, b0, (short)0, c00, false, false);
    c01 = __builtin_amdgcn_wmma_f32_16x16x4_f32(false, a0, false, b1, (short)0, c01, false, false);
    c10 = __builtin_amdgcn_wmma_f32_16x16x4_f32(false, a1, false, b0, (short)0, c10, false, false);
    c11 = __builtin_amdgcn_wmma_f32_16x16x4_f32(false, a1, false, b1, (short)0, c11, false, false);
  }

  int r0   = bm * 32 + ((lane >> 4) << 3);  // +0 or +8 within the upper 16-row tile
  int col0 = bn * 32 + m;
#pragma unroll
  for (int i = 0; i < 8; ++i) {
    C[(size_t)(r0 + i)      * Nc + col0]      = c00[i];
    C[(size_t)(r0 + i)      * Nc + col0 + 16] = c01[i];
    C[(size_t)(r0 + 16 + i) * Nc + col0]      = c10[i];
    C[(size_t)(r0 + 16 + i) * Nc + col0 + 16] = c11[i];
  }
}

// ---------- per-node attention logits, layer 1 (8 heads x 16 ch) ----------
__global__ void node_logits1(const float* __restrict__ h1, const float* __restrict__ att_s,
                             const float* __restrict__ att_d,
                             float* __restrict__ a_s, float* __restrict__ a_d, int N) {
  int t = blockIdx.x * blockDim.x + threadIdx.x;   // one per (node, head)
  if (t >= N * 8) return;
  int n = t >> 3, h = t & 7;
  const float* hp = h1 + (size_t)n * 128 + h * 16;
  const float* as = att_s + h * 16;
  const float* ad = att_d + h * 16;
  float s = 0.f, d = 0.f;
#pragma unroll
  for (int c = 0; c < 16; ++c) { float v = hp[c]; s += v * as[c]; d += v * ad[c]; }
  a_s[t] = s; a_d[t] = d;
}

// ---------- per-node attention logits, layer 2 (1 head x 32 ch) ----------
__global__ void node_logits2(const float* __restrict__ z, const float* __restrict__ att_s,
                             const float* __restrict__ att_d,
                             float* __restrict__ a_s, float* __restrict__ a_d, int N) {
  int n = blockIdx.x * blockDim.x + threadIdx.x;
  if (n >= N) return;
  const float* zp = z + (size_t)n * 32;
  float s = 0.f, d = 0.f;
#pragma unroll
  for (int c = 0; c < 32; ++c) { float v = zp[c]; s += v * att_s[c]; d += v * att_d[c]; }
  a_s[n] = s; a_d[n] = d;
}

// ---------- edge pass A: leaky-relu logits + segment max (atomicMax on encoded uint) ----------
template <int H>
__global__ void edge_logits(const int* __restrict__ src_e, const int* __restrict__ dst_e,
                            int E, int N, const float* __restrict__ a_s,
                            const float* __restrict__ a_d,
                            float* __restrict__ e_buf, unsigned* __restrict__ m_enc) {
  int e = blockIdx.x * blockDim.x + threadIdx.x;
  int EE = E + N;
  if (e >= EE) return;
  int s = (e < E) ? src_e[e] : (e - E);
  int d = (e < E) ? dst_e[e] : (e - E);
#pragma unroll
  for (int h = 0; h < H; ++h) {
    float v = a_s[s * H + h] + a_d[d * H + h];
    v = (v > 0.f) ? v : NEG_SLOPE * v;
    e_buf[(size_t)e * H + h] = v;
    atomicMax(&m_enc[d * H + h], enc_f(v));
  }
}

// ---------- edge pass B: exp(e - max) + segment sum ----------
template <int H>
__global__ void edge_exp(const int* __restrict__ dst_e, int E, int N,
                         const unsigned* __restrict__ m_enc,
                         float* __restrict__ e_buf, float* __restrict__ denom) {
  int e = blockIdx.x * blockDim.x + threadIdx.x;
  int EE = E + N;
  if (e >= EE) return;
  int d = (e < E) ? dst_e[e] : (e - E);
#pragma unroll
  for (int h = 0; h < H; ++h) {
    float m  = dec_f(m_enc[d * H + h]);
    float ex = __expf(e_buf[(size_t)e * H + h] - m);
    e_buf[(size_t)e * H + h] = ex;
    atomicAdd(&denom[d * H + h], ex);
  }
}

// ---------- edge pass C layer 1: out[dst] += h[src]*alpha (128 ch, wave per edge) ----------
__global__ void edge_scatter1(const int* __restrict__ src_e, const int* __restrict__ dst_e,
                              int E, int N, const float* __restrict__ ex,
                              const float* __restrict__ denom,
                              const float* __restrict__ h1, float* __restrict__ out) {
  int wid  = (blockIdx.x * blockDim.x + threadIdx.x) >> 5;
  int lane = threadIdx.x & 31;
  int EE = E + N;
  if (wid >= EE) return;
  int s = (wid < E) ? src_e[wid] : (wid - E);
  int d = (wid < E) ? dst_e[wid] : (wid - E);
  int h = lane >> 2;                                   // 4 lanes per head (16 ch / head)
  float alpha = ex[(size_t)wid * 8 + h] / denom[d * 8 + h];
  float4 v = *((const float4*)(h1 + (size_t)s * 128) + lane);
  float* op = out + (size_t)d * 128 + lane * 4;
  atomicAdd(op + 0, v.x * alpha);
  atomicAdd(op + 1, v.y * alpha);
  atomicAdd(op + 2, v.z * alpha);
  atomicAdd(op + 3, v.w * alpha);
}

// ---------- edge pass C layer 2: 32 ch, wave per edge, lane per channel ----------
__global__ void edge_scatter2(const int* __restrict__ src_e, const int* __restrict__ dst_e,
                              int E, int N, const float* __restrict__ ex,
                              const float* __restrict__ denom,
                              const float* __restrict__ z, float* __restrict__ out) {
  int wid  = (blockIdx.x * blockDim.x + threadIdx.x) >> 5;
  int lane = threadIdx.x & 31;
  int EE = E + N;
  if (wid >= EE) return;
  int s = (wid < E) ? src_e[wid] : (wid - E);
  int d = (wid < E) ? dst_e[wid] : (wid - E);
  float alpha = ex[wid] / denom[d];
  float v = z[(size_t)s * 32 + lane];
  atomicAdd(&out[(size_t)d * 32 + lane], v * alpha);
}

// ---------- bias + ELU (layer-1 activation), in place ----------
__global__ void bias_elu(float* __restrict__ p, const float* __restrict__ bias, long long total) {
  long long i = (long long)blockIdx.x * blockDim.x + threadIdx.x;
  if (i >= total) return;
  float v = p[i] + bias[i & 127];
  p[i] = (v > 0.f) ? v : (__expf(v) - 1.f);
}

// ---------- bias + log_softmax over 32 classes: wave per node, lane per class ----------
__global__ void logsoftmax32(const float* __restrict__ in, const float* __restrict__ bias,
                             float* __restrict__ out, int N) {
  int wid  = (blockIdx.x * blockDim.x + threadIdx.x) >> 5;
  int lane = threadIdx.x & 31;
  if (wid >= N) return;
  float v = in[(size_t)wid * 32 + lane] + bias[lane];
  float m = v;
#pragma unroll
  for (int o = 16; o > 0; o >>= 1) m = fmaxf(m, __shfl_xor(m, o, 32));
  float ex = __expf(v - m);
  float ssum = ex;
#pragma unroll
  for (int o = 16; o > 0; o >>= 1) ssum += __shfl_xor(ssum, o, 32);
  out[(size_t)wid * 32 + lane] = v - m - __logf(ssum);
}

// ========================== host launcher ==========================
extern "C" void kernel_launch(void* const* d_in, const int* in_sizes, int n_in,
                              void* d_out, int out_size, void* d_ws, size_t ws_size,
                              hipStream_t stream) {
  const float* x        = (const float*)d_in[0];
  const int*   src      = (const int*)d_in[1];
  const float* W1       = (const float*)d_in[3];
  const float* att_src1 = (const float*)d_in[4];
  const float* att_dst1 = (const float*)d_in[5];
  const float* b1       = (const float*)d_in[6];
  const float* W2       = (const float*)d_in[7];
  const float* att_src2 = (const float*)d_in[8];
  const float* att_dst2 = (const float*)d_in[9];
  const float* b2       = (const float*)d_in[10];

  const int N  = in_sizes[0] / 256;       // 100000
  const int E  = in_sizes[1] / 2;         // 1600000
  const int EE = E + N;                   // with self-loops
  const int* dst = src + E;

  // ---- workspace carve-out (all fp32 unless noted) ----
  char* ws = (char*)d_ws;
  size_t cur = 0;
  auto alloc = [&](size_t bytes) -> char* {
    char* p = ws + cur;
    cur += (bytes + 255) & ~(size_t)255;
    return p;
  };
  float*    h1     = (float*)   alloc((size_t)N * 128 * 4);
  float*    a_s1   = (float*)   alloc((size_t)N * 8 * 4);
  float*    a_d1   = (float*)   alloc((size_t)N * 8 * 4);
  unsigned* m1     = (unsigned*)alloc((size_t)N * 8 * 4);
  float*    den1   = (float*)   alloc((size_t)N * 8 * 4);
  float*    e1     = (float*)   alloc((size_t)EE * 8 * 4);
  float*    out1   = (float*)   alloc((size_t)N * 128 * 4);   // becomes h2 after bias+ELU
  float*    z2     = (float*)   alloc((size_t)N * 32 * 4);
  float*    a_s2   = (float*)   alloc((size_t)N * 4);
  float*    a_d2   = (float*)   alloc((size_t)N * 4);
  unsigned* m2     = (unsigned*)alloc((size_t)N * 4);
  float*    den2   = (float*)   alloc((size_t)N * 4);
  float*    e2     = (float*)   alloc((size_t)EE * 4);
  float*    out2   = (float*)   alloc((size_t)N * 32 * 4);
  (void)ws_size; (void)n_in; (void)out_size;

  const int T = 256;
  auto cdiv = [](long long a, long long b) { return (int)((a + b - 1) / b); };

  // ---- zero accumulators (encoded-max init 0 is overwritten: self-loops hit every dst) ----
  zero_kernel<<<2048, T, 0, stream>>>((float*)m1,   (long long)N * 8);
  zero_kernel<<<2048, T, 0, stream>>>(den1,         (long long)N * 8);
  zero_kernel<<<2048, T, 0, stream>>>(out1,         (long long)N * 128);
  zero_kernel<<<2048, T, 0, stream>>>((float*)m2,   (long long)N);
  zero_kernel<<<2048, T, 0, stream>>>(den2,         (long long)N);
  zero_kernel<<<2048, T, 0, stream>>>(out2,         (long long)N * 32);

  // ================= layer 1 =================
  {
    int waves = (N / 32) * (128 / 32);                 // 3125 * 4
    gemm_wmma_f32_2x2<<<cdiv((long long)waves * 32, T), T, 0, stream>>>(x, W1, h1, N, 256, 128);
  }
  node_logits1<<<cdiv((long long)N * 8, T), T, 0, stream>>>(h1, att_src1, att_dst1, a_s1, a_d1, N);
  edge_logits<8><<<cdiv(EE, T), T, 0, stream>>>(src, dst, E, N, a_s1, a_d1, e1, m1);
  edge_exp<8><<<cdiv(EE, T), T, 0, stream>>>(dst, E, N, m1, e1, den1);
  edge_scatter1<<<cdiv((long long)EE * 32, T), T, 0, stream>>>(src, dst, E, N, e1, den1, h1, out1);
  bias_elu<<<cdiv((long long)N * 128, T), T, 0, stream>>>(out1, b1, (long long)N * 128);

  // ================= layer 2 =================
  {
    int waves = (N / 32) * (32 / 32);                  // 3125 * 1
    gemm_wmma_f32_2x2<<<cdiv((long long)waves * 32, T), T, 0, stream>>>(out1, W2, z2, N, 128, 32);
  }
  node_logits2<<<cdiv(N, T), T, 0, stream>>>(z2, att_src2, att_dst2, a_s2, a_d2, N);
  edge_logits<1><<<cdiv(EE, T), T, 0, stream>>>(src, dst, E, N, a_s2, a_d2, e2, m2);
  edge_exp<1><<<cdiv(EE, T), T, 0, stream>>>(dst, E, N, m2, e2, den2);
  edge_scatter2<<<cdiv((long long)EE * 32, T), T, 0, stream>>>(src, dst, E, N, e2, den2, z2, out2);

  // ================= output =================
  logsoftmax32<<<cdiv((long long)N * 32, T), T, 0, stream>>>(out2, b2, (float*)d_out, N);
}